// LocalGrouper_10625749090927
// MI455X (gfx1250) — compile-verified
//
#include <hip/hip_runtime.h>
#include <hip/hip_bf16.h>
#include <math.h>

#define BB 16
#define NN 4096
#define SS 1024
#define KK 32
#define CC 64
#define CH 67      // C + 3
#define OC 131     // (C+3) + C
#define EPSV 1e-5f

typedef __attribute__((ext_vector_type(2))) float v2f;
typedef __attribute__((ext_vector_type(8))) float v8f;

#define CHUNK 512
#define WAVES 4
#define TPB   (WAVES * 32)   // 128 threads
#define QPW   16             // queries per wave (one WMMA N-tile)
#define QPB   (WAVES * QPW)  // 64 queries per block

// ---------------------------------------------------------------------------
// Kernel 1: KNN via V_WMMA_F32_16X16X4_F32.
// score(p,q) = -2 p.q + |p|^2  (|q|^2 dropped: constant per query row).
// A (16x4) = point rows (x,y,z,|p|^2); B (4x16) = query cols (-2x,-2y,-2z,1).
// D layout: lane owns query column n=lane%16; VGPR j = point row j (+8 for
// lanes 16-31). Each lane keeps a sorted top-32 over its half of the points
// in registers; halves merged per query at the end through LDS.
// The A-tile ds_load is software-pipelined one tile ahead so LDS latency
// overlaps the WMMA + compare work instead of stalling on s_wait_dscnt.
// ---------------------------------------------------------------------------
__global__ __launch_bounds__(TPB) void knn_kernel(
    const float* __restrict__ xyz, const float* __restrict__ new_xyz,
    int* __restrict__ knn)
{
  __shared__ float4 s_pts[CHUNK];            // staged points: x,y,z,|p|^2
  __shared__ float  s_md[WAVES][32][KK];     // per-lane sorted dists
  __shared__ int    s_mi[WAVES][32][KK];     // per-lane sorted indices

  const int b    = blockIdx.y;
  const int tid  = threadIdx.x;
  const int wave = tid >> 5;
  const int lane = tid & 31;
  const int lq   = lane & 15;
  const bool lo  = (lane < 16);

  // B operand (queries): constant across the whole N scan.
  const int q = blockIdx.x * QPB + wave * QPW + lq;
  const size_t qoff = ((size_t)b * SS + q) * 3;
  const float qx = new_xyz[qoff + 0];
  const float qy = new_xyz[qoff + 1];
  const float qz = new_xyz[qoff + 2];
  v2f bop;
  bop.x = lo ? (-2.0f * qx) : (-2.0f * qz);
  bop.y = lo ? (-2.0f * qy) : 1.0f;

  float dist[KK];
  int   idx[KK];
#pragma unroll
  for (int j = 0; j < KK; ++j) { dist[j] = 3.4e38f; idx[j] = 0; }

  for (int c0 = 0; c0 < NN; c0 += CHUNK) {
    __syncthreads();
    for (int i = tid; i < CHUNK; i += TPB) {
      const size_t p = ((size_t)b * NN + c0 + i) * 3;
      float x = xyz[p + 0], y = xyz[p + 1], z = xyz[p + 2];
      s_pts[i] = make_float4(x, y, z, x * x + y * y + z * z);
    }
    __syncthreads();

    float4 P = s_pts[lq];                       // prime the pipeline
    for (int t = 0; t < CHUNK; t += 16) {
      // issue next tile's load now; wrap-around keeps it branchless
      float4 Pn = s_pts[((t + 16) & (CHUNK - 1)) + lq];

      v2f aop;                                  // A tile: 16 points x 4
      aop.x = lo ? P.x : P.z;
      aop.y = lo ? P.y : P.w;
      v8f acc = {};
      acc = __builtin_amdgcn_wmma_f32_16x16x4_f32(
          false, aop, false, bop, (short)0, acc, false, false);
#pragma unroll
      for (int j = 0; j < 8; ++j) {
        float d = acc[j];
        int pi = c0 + t + j + (lo ? 0 : 8);
        if (d < dist[KK - 1]) {                 // rare after warm-up
#pragma unroll
          for (int m = KK - 1; m >= 1; --m) {   // static idx -> stays in VGPRs
            bool sh = dist[m - 1] > d;
            bool pl = (!sh) && (dist[m] > d);
            dist[m] = sh ? dist[m - 1] : (pl ? d  : dist[m]);
            idx[m]  = sh ? idx[m - 1]  : (pl ? pi : idx[m]);
          }
          if (dist[0] > d) { dist[0] = d; idx[0] = pi; }
        }
      }
      P = Pn;
    }
  }

  // Dump both half-lists, then lanes 0-15 merge (two-pointer over sorted lists)
#pragma unroll
  for (int j = 0; j < KK; ++j) {
    s_md[wave][lane][j] = dist[j];
    s_mi[wave][lane][j] = idx[j];
  }
  __syncthreads();

  if (lo) {
    int ia = 0, ib = 0;
    int* out = knn + ((size_t)b * SS + q) * KK;
    for (int j = 0; j < KK; ++j) {
      float da = s_md[wave][lane][ia];
      float db = s_md[wave][lane + 16][ib];
      bool ta = (da <= db);
      out[j] = ta ? s_mi[wave][lane][ia] : s_mi[wave][lane + 16][ib];
      ia += ta ? 1 : 0;
      ib += ta ? 0 : 1;
    }
  }
}

// ---------------------------------------------------------------------------
// Kernel 2: per-group mean (stored) and per-group sum of squared deviations.
// One block per (b,s); thread t = channel (t<67 active).
// ---------------------------------------------------------------------------
#define STB 96
__global__ __launch_bounds__(STB) void stats_kernel(
    const float* __restrict__ xyz, const float* __restrict__ features,
    const int* __restrict__ knn, float* __restrict__ mean_out,
    float* __restrict__ gs_out)
{
  __shared__ int   s_idx[KK];
  __shared__ float s_red[STB];
  const int b = blockIdx.y, s = blockIdx.x, t = threadIdx.x;
  if (t < KK) s_idx[t] = knn[((size_t)b * SS + s) * KK + t];
  __syncthreads();

  float devsq = 0.0f;
  if (t < CH) {
    float sum = 0.0f, sq = 0.0f;
#pragma unroll 4
    for (int k = 0; k < KK; ++k) {
      int p = s_idx[k];
      float v = (t < CC) ? features[((size_t)b * NN + p) * CC + t]
                         : xyz[((size_t)b * NN + p) * 3 + (t - CC)];
      sum += v;
      sq  += v * v;
    }
    float m = sum * (1.0f / KK);
    mean_out[((size_t)b * SS + s) * CH + t] = m;
    devsq = sq - (float)KK * m * m;           // sum (v-m)^2
    if (devsq < 0.0f) devsq = 0.0f;
  }
  s_red[t] = devsq;
  __syncthreads();
  if (t == 0) {
    float tot = 0.0f;
    for (int i = 0; i < STB; ++i) tot += s_red[i];
    gs_out[(size_t)b * SS + s] = tot;         // no atomics: deterministic
  }
}

// ---------------------------------------------------------------------------
// Kernel 3: per-batch reduction -> inv_std = 1 / (sqrt(sum/(cnt-1)) + eps)
// ---------------------------------------------------------------------------
__global__ __launch_bounds__(256) void reduce_kernel(
    const float* __restrict__ gs, float* __restrict__ istd)
{
  __shared__ float s_red[256];
  const int b = blockIdx.x, t = threadIdx.x;
  float local = 0.0f;
  for (int s = t; s < SS; s += 256) local += gs[(size_t)b * SS + s];
  s_red[t] = local;
  __syncthreads();
  for (int off = 128; off > 0; off >>= 1) {
    if (t < off) s_red[t] += s_red[t + off];
    __syncthreads();
  }
  if (t == 0) {
    const float cnt = (float)SS * (float)KK * (float)CH;
    float var = s_red[0] / (cnt - 1.0f);
    istd[b] = 1.0f / (sqrtf(var) + EPSV);
  }
}

// ---------------------------------------------------------------------------
// Kernel 4: write output [B,S,K,131]
// ---------------------------------------------------------------------------
__global__ __launch_bounds__(256) void finalize_kernel(
    const float* __restrict__ xyz, const float* __restrict__ features,
    const float* __restrict__ new_features,
    const float* __restrict__ alpha, const float* __restrict__ beta,
    const int* __restrict__ knn, const float* __restrict__ mean,
    const float* __restrict__ istd, float* __restrict__ out, long long total)
{
  long long i = (long long)blockIdx.x * blockDim.x + threadIdx.x;
  if (i >= total) return;
  int c = (int)(i % OC);
  long long g = i / OC;            // flattened (b,s,k)
  int k = (int)(g % KK);
  long long bs = g / KK;           // b*S + s
  int b = (int)(bs / SS);
  float r;
  if (c < CH) {
    int p = knn[bs * KK + k];
    float v = (c < CC) ? features[((size_t)b * NN + p) * CC + c]
                       : xyz[((size_t)b * NN + p) * 3 + (c - CC)];
    float m = mean[bs * CH + c];
    r = alpha[c] * ((v - m) * istd[b]) + beta[c];
  } else {
    r = new_features[bs * CC + (c - CH)];
  }
  out[i] = r;
}

// ---------------------------------------------------------------------------
extern "C" void kernel_launch(void* const* d_in, const int* in_sizes, int n_in,
                              void* d_out, int out_size, void* d_ws, size_t ws_size,
                              hipStream_t stream) {
  const float* xyz          = (const float*)d_in[0];
  const float* features     = (const float*)d_in[1];
  const float* new_xyz      = (const float*)d_in[2];
  const float* new_features = (const float*)d_in[3];
  const float* alpha        = (const float*)d_in[4];
  const float* beta         = (const float*)d_in[5];
  float* out = (float*)d_out;

  char* ws = (char*)d_ws;
  size_t off = 0;
  int*   knn  = (int*)(ws + off);   off += (size_t)BB * SS * KK * sizeof(int);
  float* mean = (float*)(ws + off); off += (size_t)BB * SS * CH * sizeof(float);
  float* gs   = (float*)(ws + off); off += (size_t)BB * SS * sizeof(float);
  float* istd = (float*)(ws + off);

  knn_kernel<<<dim3(SS / QPB, BB), TPB, 0, stream>>>(xyz, new_xyz, knn);
  stats_kernel<<<dim3(SS, BB), STB, 0, stream>>>(xyz, features, knn, mean, gs);
  reduce_kernel<<<BB, 256, 0, stream>>>(gs, istd);

  const long long total = (long long)BB * SS * KK * OC;
  const int blocks = (int)((total + 255) / 256);
  finalize_kernel<<<blocks, 256, 0, stream>>>(xyz, features, new_features,
                                              alpha, beta, knn, mean, istd,
                                              out, total);
}